// RelGraphConv_49289044689304
// MI455X (gfx1250) — compile-verified
//
#include <hip/hip_runtime.h>

typedef float v2f __attribute__((ext_vector_type(2)));
typedef float v8f __attribute__((ext_vector_type(8)));

#define IN_FEAT   64
#define OUT_FEAT  64
#define NUM_BASES 4
#define ZW        (NUM_BASES * OUT_FEAT)   // 256 floats per node in workspace

// ---------------------------------------------------------------------------
// Kernel 1: node-level GEMM via V_WMMA_F32_16X16X4_F32.
// Virtual B matrix = [V0 | V1 | V2 | V3 | loop_weight]  (64 x 320).
// j (col tile, 0..19): j<16 -> z[n][j*16..] into workspace,
//                      j>=16 -> h[n][(j-16)*16..] = x@loop_weight + bias.
// One wave computes one 16x16 output tile, 16 chained WMMAs over K=64.
// ---------------------------------------------------------------------------
__global__ __launch_bounds__(128) void rgcn_node_gemm(
    const float* __restrict__ x,            // (N, 64)
    const float* __restrict__ weight,       // (4, 64, 64)
    const float* __restrict__ loop_weight,  // (64, 64)
    const float* __restrict__ h_bias,       // (64)
    float* __restrict__ z,                  // ws: (N, 256)
    float* __restrict__ h,                  // out: (N, 64)
    int n_nodes)
{
    const int lane = threadIdx.x & 31;
    const int lm   = lane & 15;
    const int half = lane >> 4;                       // 0: K lo pair, 1: K hi pair
    const int j    = blockIdx.y * blockDim.y + threadIdx.y; // 0..19, wave-uniform
    const int m0   = blockIdx.x * 16;

    int row = m0 + lm;
    if (row >= n_nodes) row = n_nodes - 1;            // clamp: keep EXEC all-ones

    const float* Bmat;
    int colBase;
    if (j < 16) {                                     // basis b = j>>2
        Bmat    = weight + (size_t)(j >> 2) * (IN_FEAT * OUT_FEAT);
        colBase = (j & 3) * 16;
    } else {                                          // self-loop block
        Bmat    = loop_weight;
        colBase = (j - 16) * 16;
    }
    const int col = colBase + lm;

    const float* arow = x + (size_t)row * IN_FEAT;
    v8f acc = {};
#pragma unroll
    for (int kc = 0; kc < 16; ++kc) {
        const int k = kc * 4 + half * 2;
        // A 16x4 f32: lane m holds K={k,k+1} -> contiguous 8B load
        v2f a = *(const v2f*)(arow + k);
        // B 4x16 f32: lane n holds rows K={k,k+1} of column `col`
        v2f b;
        b.x = Bmat[(size_t)(k + 0) * OUT_FEAT + col];
        b.y = Bmat[(size_t)(k + 1) * OUT_FEAT + col];
        acc = __builtin_amdgcn_wmma_f32_16x16x4_f32(
                  false, a, false, b, (short)0, acc, false, false);
    }

    // C/D layout: VGPR r, lanes 0-15 -> M=r, lanes 16-31 -> M=r+8, N=lane%16
    if (j < 16) {
#pragma unroll
        for (int r = 0; r < 8; ++r) {
            const int m = m0 + r + half * 8;
            if (m < n_nodes)
                z[(size_t)m * ZW + j * 16 + lm] = acc[r];
        }
    } else {
        const float bias = h_bias[col];
#pragma unroll
        for (int r = 0; r < 8; ++r) {
            const int m = m0 + r + half * 8;
            if (m < n_nodes)
                h[(size_t)m * OUT_FEAT + col] = acc[r] + bias;
        }
    }
}

// ---------------------------------------------------------------------------
// Kernel 2: one wave per edge.
//   msg = norm[e] * sum_b w_comp[etype[e]][b] * z[src[e]][b*64 + o]
//   h[dst[e]][o] += msg   via native global_atomic_add_f32 (L2-resident)
// Lane l covers o = l and o = l+32: each atomic instruction is a contiguous
// 128B span across the wave (2 cachelines), 4 cacheline txns per edge total.
// ---------------------------------------------------------------------------
__global__ __launch_bounds__(256) void rgcn_edge(
    const float* __restrict__ z,        // (N, 256)
    const float* __restrict__ w_comp,   // (64, 4)
    const float* __restrict__ norm,     // (E, 1)
    const int*   __restrict__ src,
    const int*   __restrict__ dst,
    const int*   __restrict__ et,
    float* __restrict__ h,              // (N, 64)
    int n_edges)
{
    const int lane = threadIdx.x & 31;
    const int e    = blockIdx.x * (blockDim.x >> 5) + (threadIdx.x >> 5);
    if (e >= n_edges) return;

    const int   s  = src[e];
    const int   d  = dst[e];
    const int   r  = et[e];
    const float nm = norm[e];
    const float4 cf = *(const float4*)(w_comp + (size_t)r * NUM_BASES);

    const float* zp = z + (size_t)s * ZW + lane;      // o = lane, lane+32
    const float z0a = zp[0 * OUT_FEAT];      const float z0b = zp[0 * OUT_FEAT + 32];
    const float z1a = zp[1 * OUT_FEAT];      const float z1b = zp[1 * OUT_FEAT + 32];
    const float z2a = zp[2 * OUT_FEAT];      const float z2b = zp[2 * OUT_FEAT + 32];
    const float z3a = zp[3 * OUT_FEAT];      const float z3b = zp[3 * OUT_FEAT + 32];

    const float ma = nm * (cf.x * z0a + cf.y * z1a + cf.z * z2a + cf.w * z3a);
    const float mb = nm * (cf.x * z0b + cf.y * z1b + cf.z * z2b + cf.w * z3b);

    float* hp = h + (size_t)d * OUT_FEAT + lane;
    unsafeAtomicAdd(hp + 0,  ma);
    unsafeAtomicAdd(hp + 32, mb);
}

// ---------------------------------------------------------------------------
// Launch: GEMM (writes z to ws, initializes d_out with selfloop+bias), then
// edge scatter kernel accumulating into d_out. Same stream -> ordered.
// ---------------------------------------------------------------------------
extern "C" void kernel_launch(void* const* d_in, const int* in_sizes, int n_in,
                              void* d_out, int out_size, void* d_ws, size_t ws_size,
                              hipStream_t stream)
{
    const float* x           = (const float*)d_in[0];
    const float* weight      = (const float*)d_in[1];
    const float* w_comp      = (const float*)d_in[2];
    const float* h_bias      = (const float*)d_in[3];
    const float* loop_weight = (const float*)d_in[4];
    const float* norm        = (const float*)d_in[5];
    const int*   src         = (const int*)d_in[6];
    const int*   dst         = (const int*)d_in[7];
    const int*   etypes      = (const int*)d_in[8];

    const int n_nodes = in_sizes[0] / IN_FEAT;   // 50000
    const int n_edges = in_sizes[6];             // 800000

    float* z = (float*)d_ws;                     // needs N*256*4 = 51.2 MB
    float* h = (float*)d_out;

    // Kernel 1: 4 waves/block (threadIdx.y = col-tile within group of 4),
    // 20 col tiles total -> grid.y = 5; grid.x = node tiles of 16.
    dim3 b1(32, 4);
    dim3 g1((n_nodes + 15) / 16, 5);
    rgcn_node_gemm<<<g1, b1, 0, stream>>>(x, weight, loop_weight, h_bias,
                                          z, h, n_nodes);

    // Kernel 2: 8 waves (edges) per 256-thread block.
    const int wpb = 256 / 32;
    const int g2  = (n_edges + wpb - 1) / wpb;
    rgcn_edge<<<g2, 256, 0, stream>>>(z, w_comp, norm, src, dst, etypes,
                                      h, n_edges);
}